// DiffPairRandomRotate_35141422416279
// MI455X (gfx1250) — compile-verified
//
#include <hip/hip_runtime.h>
#include <math.h>

// DiffPairRandomRotate for MI455X (gfx1250, wave32).
// Memory-bound bilinear gather; staged through LDS with the CDNA5 async
// global->LDS copy path (ASYNCcnt), double-buffered over the 16-channel loop.
// Staging uses the GVS (scalar-base + 32-bit vector offset) addressing mode
// with channel-invariant offsets hoisted out of the loop.

namespace {
constexpr int Bn = 4, Cn = 8, Hn = 512, Wn = 512;
constexpr int PH = 107, PW = 107;           // (int(sqrt2*512)-512)//2 + 1
constexpr int HP = Hn + 2 * PH;             // 726
constexpr int WP = Wn + 2 * PW;             // 726
constexpr int TX = 32, TY = 8;              // output tile per workgroup
constexpr int LW = 36, LH = 36;             // LDS staging tile (covers rotated footprint)
constexpr int LSZ = LW * LH;                // 1296 elements
constexpr int NT  = 256;                    // 8 waves of 32
constexpr int NK  = (LSZ + NT - 1) / NT;    // 6 staging iterations
constexpr int NCH = 2 * Cn;                 // 16 channels (x then y)
constexpr size_t PLANE = (size_t)Hn * Wn;
constexpr size_t BCHW  = (size_t)Bn * Cn * Hn * Wn;
}

__device__ __forceinline__ void map_coord(float jx, float jy, float cs, float sn,
                                          float& ix, float& iy) {
    // padded pixel (jx,jy) -> normalized grid -> rotated -> padded sample pos
    float px = fmaf(jx, 2.0f / (float)(WP - 1), -1.0f);
    float py = fmaf(jy, 2.0f / (float)(HP - 1), -1.0f);
    float gx = px * cs - py * sn;
    float gy = px * sn + py * cs;
    ix = ((gx + 1.0f) * (float)WP - 1.0f) * 0.5f;
    iy = ((gy + 1.0f) * (float)HP - 1.0f) * 0.5f;
}

__global__ __launch_bounds__(NT)
void rotate_pair_sample_kernel(const float* __restrict__ X,
                               const float* __restrict__ Y,
                               const float* __restrict__ ANG,
                               float* __restrict__ OUT)
{
    __shared__ float smem[2][LSZ];

    const int tid   = threadIdx.x;
    const int b     = blockIdx.y;
    const int tileX = blockIdx.x & (Wn / TX - 1);   // 16 tiles across
    const int tileY = blockIdx.x >> 4;              // 64 tiles down
    const int ox0   = tileX * TX;
    const int oy0   = tileY * TY;

    // |rad| <= pi/4 -> hardware sin/cos path is plenty accurate, no libm slow path
    const float rad = ANG[b] * 0.017453292519943295f;
    float sn, cs;
    __sincosf(rad, &sn, &cs);

    // ---- workgroup-uniform bounding box of this tile's sample footprint ----
    float ixA, iyA, ixB, iyB, ixC, iyC, ixD, iyD;
    map_coord((float)(ox0 + PW),          (float)(oy0 + PH),          cs, sn, ixA, iyA);
    map_coord((float)(ox0 + TX - 1 + PW), (float)(oy0 + PH),          cs, sn, ixB, iyB);
    map_coord((float)(ox0 + PW),          (float)(oy0 + TY - 1 + PH), cs, sn, ixC, iyC);
    map_coord((float)(ox0 + TX - 1 + PW), (float)(oy0 + TY - 1 + PH), cs, sn, ixD, iyD);
    const int bx0 = (int)floorf(fminf(fminf(ixA, ixB), fminf(ixC, ixD))) - 1;
    const int by0 = (int)floorf(fminf(fminf(iyA, iyB), fminf(iyC, iyD))) - 1;

    // ---- per-thread sample position, weights, masks (amortized over 16 ch) ----
    const int tx = tid & (TX - 1);
    const int ty = tid >> 5;
    const int ox = ox0 + tx;
    const int oy = oy0 + ty;

    float ix, iy;
    map_coord((float)(ox + PW), (float)(oy + PH), cs, sn, ix, iy);
    const float x0f = floorf(ix), y0f = floorf(iy);
    const int   x0  = (int)x0f,   y0  = (int)y0f;
    const float wx1 = ix - x0f, wx0 = 1.0f - wx1;
    const float wy1 = iy - y0f, wy0 = 1.0f - wy1;

    // zero-padding + zeros grid_sample collapse into an in-original-image mask
    const int ax0 = x0 - PW, ay0 = y0 - PH;
    const bool vx0 = (ax0 >= 0)     && (ax0 < Wn);
    const bool vx1 = (ax0 + 1 >= 0) && (ax0 + 1 < Wn);
    const bool vy0 = (ay0 >= 0)     && (ay0 < Hn);
    const bool vy1 = (ay0 + 1 >= 0) && (ay0 + 1 < Hn);
    const float w00 = wx0 * wy0 * ((vx0 && vy0) ? 1.0f : 0.0f);
    const float w01 = wx1 * wy0 * ((vx1 && vy0) ? 1.0f : 0.0f);
    const float w10 = wx0 * wy1 * ((vx0 && vy1) ? 1.0f : 0.0f);
    const float w11 = wx1 * wy1 * ((vx1 && vy1) ? 1.0f : 0.0f);

    int lx0 = x0 - bx0; lx0 = lx0 < 0 ? 0 : (lx0 > LW - 2 ? LW - 2 : lx0);
    int ly0 = y0 - by0; ly0 = ly0 < 0 ? 0 : (ly0 > LH - 2 ? LH - 2 : ly0);
    const int base0 = ly0 * LW + lx0;

    // ---- channel-invariant staging offsets (computed once) ----
    // boff[k]: per-lane byte offset into the source plane (clamped coords;
    //          out-of-image samples are zero-masked in the weights above)
    // loff[k]: per-lane byte offset into an LDS buffer
    int      boff[NK];
    unsigned loff[NK];
#pragma unroll
    for (int k = 0; k < NK; ++k) {
        int i = tid + k * NT;
        if (i > LSZ - 1) i = LSZ - 1;       // duplicate tail element (benign)
        const int ly = i / LW;
        const int lx = i - ly * LW;
        int ayy = by0 + ly - PH;
        int axx = bx0 + lx - PW;
        ayy = ayy < 0 ? 0 : (ayy > Hn - 1 ? Hn - 1 : ayy);
        axx = axx < 0 ? 0 : (axx > Wn - 1 ? Wn - 1 : axx);
        boff[k] = (ayy * Wn + axx) * 4;     // < 1 MiB, fits in i32
        loff[k] = (unsigned)i * 4u;
    }
    const unsigned sbase0 = (unsigned)(size_t)(const void*)&smem[0][0];

    // ---- async stage one channel's footprint into LDS buffer s (GVS mode) ----
    auto stage = [&](int c, int s) {
        const float* plane = ((c < Cn) ? X : Y) + ((size_t)b * Cn + (c & (Cn - 1))) * PLANE;
        const unsigned sb = sbase0 + (unsigned)s * (unsigned)(LSZ * 4);
#pragma unroll
        for (int k = 0; k < NK; ++k) {
            const unsigned la = sb + loff[k];
            asm volatile("global_load_async_to_lds_b32 %0, %1, %2"
                         :: "v"(la), "v"(boff[k]), "s"(plane) : "memory");
        }
    };

    stage(0, 0);
    asm volatile("s_wait_asynccnt 0x0" ::: "memory");
    __syncthreads();

    const size_t out_pix = (size_t)oy * Wn + ox;

    for (int c = 0; c < NCH; ++c) {
        const int s = c & 1;
        if (c + 1 < NCH) stage(c + 1, s ^ 1);   // overlap next channel's copy

        const float v00 = smem[s][base0];
        const float v01 = smem[s][base0 + 1];
        const float v10 = smem[s][base0 + LW];
        const float v11 = smem[s][base0 + LW + 1];
        const float r = fmaf(v00, w00, fmaf(v01, w01, fmaf(v10, w10, v11 * w11)));

        float* dst = OUT + ((c < Cn) ? (size_t)0 : BCHW)
                   + ((size_t)b * Cn + (c & (Cn - 1))) * PLANE + out_pix;
        __builtin_nontemporal_store(r, dst);    // write-once output, keep L2 for gathers

        if (c + 1 < NCH) {
            asm volatile("s_wait_asynccnt 0x0" ::: "memory");
            __syncthreads();
        }
    }
}

extern "C" void kernel_launch(void* const* d_in, const int* in_sizes, int n_in,
                              void* d_out, int out_size, void* d_ws, size_t ws_size,
                              hipStream_t stream) {
    (void)in_sizes; (void)n_in; (void)out_size; (void)d_ws; (void)ws_size;
    const float* X   = (const float*)d_in[0];
    const float* Y   = (const float*)d_in[1];
    const float* ANG = (const float*)d_in[2];
    float* OUT = (float*)d_out;

    dim3 grid((Wn / TX) * (Hn / TY), Bn, 1);   // 1024 tiles x 4 batches
    rotate_pair_sample_kernel<<<grid, NT, 0, stream>>>(X, Y, ANG, OUT);
}